// GGSUp_1090921693815
// MI455X (gfx1250) — compile-verified
//
#include <hip/hip_runtime.h>
#include <hip/hip_bf16.h>
#include <cstdint>
#include <math.h>

// Problem constants (from reference setup_inputs): x [2,256,128,128] f32
#define BB 2
#define CC 256
#define HH 128
#define WW 128
#define H2 (2*HH)
#define W2 (2*WW)
#define HWH (HH*WW)

typedef __attribute__((ext_vector_type(2))) float v2f;
typedef __attribute__((ext_vector_type(8))) float v8f;

// ---------------------------------------------------------------------------
// Stage 1: mean over channels via V_WMMA_F32_16X16X4_F32.
// One wave handles 16 consecutive spatial positions (M dim). K steps over
// channels 4 at a time. B matrix is all-ones so D[m,n] = sum_k A[m,k] for
// every n, independent of the (undocumented) B VGPR layout.
// A 16x4 f32 layout: lanes 0-15 -> M=lane, VGPR0/1 = K0/K1;
//                    lanes 16-31 -> M=lane-16, VGPR0/1 = K2/K3.
// D 16x16 f32 layout: lane l, VGPR v -> M = v + 8*(l>=16), N = l&15.
// ---------------------------------------------------------------------------
__global__ __launch_bounds__(256) void k_mean(const float* __restrict__ x,
                                              float* __restrict__ mean) {
  int lane = threadIdx.x & 31;
  int wave = (int)((blockIdx.x * blockDim.x + threadIdx.x) >> 5); // 0..2047
  int sg = wave << 4;           // spatial group base over b*HWH (16384 | 16)
  int b  = sg >> 14;            // /16384
  int s  = sg & (HWH - 1);
#if __has_builtin(__builtin_amdgcn_wmma_f32_16x16x4_f32)
  int m  = lane & 15;
  int kh = (lane >> 4) << 1;    // 0 or 2
  const float* px = x + (size_t)b * CC * HWH + s + m;
  v8f acc = {};
  v2f bv;  bv[0] = 1.0f; bv[1] = 1.0f;
  for (int c = 0; c < CC; c += 4) {
    v2f av;
    av[0] = px[(size_t)(c + kh)     * HWH];
    av[1] = px[(size_t)(c + kh + 1) * HWH];
    acc = __builtin_amdgcn_wmma_f32_16x16x4_f32(
        false, av, false, bv, (short)0, acc, false, false);
  }
  if (m == 0) {                 // lanes 0 and 16 hold column N=0 (M=0..7 / 8..15)
    int mb = (lane >> 4) << 3;
    float* po = mean + b * HWH + s + mb;
#pragma unroll
    for (int v = 0; v < 8; ++v) po[v] = acc[v] * (1.0f / 256.0f);
  }
#else
  if (lane < 16) {
    int s2 = s + lane;
    const float* px = x + (size_t)b * CC * HWH + s2;
    float acc = 0.f;
    for (int c = 0; c < CC; ++c) acc += px[(size_t)c * HWH];
    mean[b * HWH + s2] = acc * (1.0f / 256.0f);
  }
#endif
}

// ---------------------------------------------------------------------------
// Stage 2: per output pixel (b,y2,x2) compute the 9 softmax weights.
// The 5x5 conv on nearest-up2(mean) only touches mean rows/cols h-1..h+1,
// the same 3x3 block the gradient gating needs.
// Weights stored planar: wk[((b*9+k)*H2+y2)*W2+x2] (coalesced in stage 3).
// ---------------------------------------------------------------------------
__global__ __launch_bounds__(256) void k_weights(const float* __restrict__ mean,
                                                 const float* __restrict__ Woff,
                                                 const float* __restrict__ boff,
                                                 float* __restrict__ wk) {
  int t  = (int)(blockIdx.x * blockDim.x + threadIdx.x);  // 0..131071
  int b  = t >> 16;
  int yx = t & 65535;
  int y2 = yx >> 8;
  int x2 = yx & 255;
  int h = y2 >> 1, w = x2 >> 1;

  float m3[3][3];
#pragma unroll
  for (int di = 0; di < 3; ++di)
#pragma unroll
    for (int dj = 0; dj < 3; ++dj) {
      int hh = h - 1 + di, wc = w - 1 + dj;
      m3[di][dj] = (hh >= 0 && hh < HH && wc >= 0 && wc < WW)
                       ? mean[b * HWH + hh * WW + wc] : 0.0f;
    }

  // conv5x5 in upsampled space, padding 2 (zero pad in up-space == mean range)
  float off0 = 0.f, off1 = 0.f;
#pragma unroll
  for (int u = 0; u < 5; ++u) {
    int yy  = y2 + u - 2;
    bool yok = (yy >= 0 && yy < H2);
    int ri  = (yy >> 1) - (h - 1);            // 0..2 when yok
#pragma unroll
    for (int v = 0; v < 5; ++v) {
      int xx = x2 + v - 2;
      float mu = 0.f;
      if (yok && xx >= 0 && xx < W2) mu = m3[ri][(xx >> 1) - (w - 1)];
      off0 += mu * Woff[u * 5 + v];
      off1 += mu * Woff[25 + u * 5 + v];
    }
  }
  float sh0 = ((x2 & 1) ? 0.25f : -0.25f) + 0.25f * tanhf(off0 + boff[0]);
  float sh1 = ((y2 & 1) ? 0.25f : -0.25f) + 0.25f * tanhf(off1 + boff[1]);

  float logit[9];
  float mx = -1e30f;
#pragma unroll
  for (int k = 0; k < 9; ++k) {
    int ki = k / 3, kj = k % 3;
    float d0 = sh0 - (float)(kj - 1);        // n0 fast over width
    float d1 = sh1 - (float)(ki - 1);        // n1 slow over height
    float kern = 1.0f / (d0 * d0 + d1 * d1 + 0.2f);
    float gr = m3[ki][kj] - m3[1][1];
    float g  = 1.0f / (gr * gr + 1.0f);
    logit[k] = g * kern;
    mx = fmaxf(mx, logit[k]);
  }
  float sum = 0.f;
#pragma unroll
  for (int k = 0; k < 9; ++k) { logit[k] = expf(logit[k] - mx); sum += logit[k]; }
  float inv = 1.0f / sum;
#pragma unroll
  for (int k = 0; k < 9; ++k)
    wk[(((size_t)b * 9 + k) * H2 + y2) * W2 + x2] = logit[k] * inv;
}

// ---------------------------------------------------------------------------
// CDNA5 async global->LDS copy (ASYNCcnt path), portable via inline asm.
// Generic->LDS address: low 32 bits of a generic LDS pointer are the LDS
// byte address.
// ---------------------------------------------------------------------------
__device__ __forceinline__ void async_ld_f32(void* lds, const float* g) {
  unsigned la = (unsigned)(unsigned long long)(uintptr_t)lds;
  unsigned long long ga = (unsigned long long)(uintptr_t)g;
  asm volatile("global_load_async_to_lds_b32 %0, %1, off"
               :: "v"(la), "v"(ga) : "memory");
}
__device__ __forceinline__ void wait_async0() {
#if __has_builtin(__builtin_amdgcn_s_wait_asynccnt)
  __builtin_amdgcn_s_wait_asynccnt(0);
#else
  asm volatile("s_wait_asynccnt 0" ::: "memory");
#endif
}

// ---------------------------------------------------------------------------
// Stage 3: CARAFE reassembly. Block = 256 threads = 32x8 output tile.
// Input tile per channel: 6 rows x 18 cols, double-buffered in LDS via async
// copies. Weights (9 per thread) live in registers for all 256 channels.
// Output stores: 32 consecutive floats per wave row = 128B coalesced.
// ---------------------------------------------------------------------------
#define TLW 32
#define TLH 8
#define ITH (TLH / 2 + 2)   // 6
#define ITW (TLW / 2 + 2)   // 18
#define NIT (ITH * ITW)     // 108

__global__ __launch_bounds__(256) void k_carafe(const float* __restrict__ x,
                                                const float* __restrict__ wk,
                                                float* __restrict__ out) {
  __shared__ float tile[2][ITH][ITW];
  int t  = threadIdx.x;
  int tx = t & 31, ty = t >> 5;
  int b  = blockIdx.z;
  int y2 = blockIdx.y * TLH + ty;
  int x2 = blockIdx.x * TLW + tx;
  int hb = blockIdx.y * (TLH / 2);
  int wb = blockIdx.x * (TLW / 2);

  float wgt[9];
#pragma unroll
  for (int k = 0; k < 9; ++k)
    wgt[k] = wk[(((size_t)b * 9 + k) * H2 + y2) * W2 + x2];

  int lr = ty >> 1;            // 0..3  (tile row 0 == input row hb-1)
  int lc = tx >> 1;            // 0..15 (tile col 0 == input col wb-1)

  bool loader = (t < NIT);
  int r  = t / ITW, cc = t % ITW;
  int gh = hb - 1 + r, gw = wb - 1 + cc;
  bool inr = loader && gh >= 0 && gh < HH && gw >= 0 && gw < WW;
  if (loader && !inr) {        // channel-invariant zeros, written once
    tile[0][r][cc] = 0.f;
    tile[1][r][cc] = 0.f;
  }
  const float* gbase =
      x + (size_t)b * CC * HWH + (size_t)(inr ? gh : 0) * WW + (inr ? gw : 0);
  float* obase = out + (((size_t)b * CC) * H2 + y2) * W2 + x2;

  if (inr) async_ld_f32(&tile[0][r][cc], gbase);   // prologue: channel 0

  for (int c = 0; c < CC; ++c) {
    int cur = c & 1;
    wait_async0();             // my wave's async writes into tile[cur] landed
    __syncthreads();           // everyone's landed; prev channel fully read
    if (c + 1 < CC && inr)
      async_ld_f32(&tile[cur ^ 1][r][cc], gbase + (size_t)(c + 1) * HWH);
    float s = 0.f;
#pragma unroll
    for (int k = 0; k < 9; ++k)
      s += wgt[k] * tile[cur][lr + k / 3][lc + k % 3];
    obase[(size_t)c * (H2 * W2)] = s;
  }
}

// ---------------------------------------------------------------------------
extern "C" void kernel_launch(void* const* d_in, const int* in_sizes, int n_in,
                              void* d_out, int out_size, void* d_ws, size_t ws_size,
                              hipStream_t stream) {
  const float* x    = (const float*)d_in[0];
  const float* Woff = (const float*)d_in[1];
  const float* boff = (const float*)d_in[2];
  float* out  = (float*)d_out;
  float* mean = (float*)d_ws;                 // 2*128*128 floats   (128 KiB)
  float* wk   = mean + BB * HWH;              // 2*9*256*256 floats (4.5 MiB)

  // Stage 1: 2048 full waves -> 256 blocks x 256 threads (EXEC all-1s for WMMA)
  k_mean<<<256, 256, 0, stream>>>(x, mean);
  // Stage 2: 2*256*256 pixels -> 512 blocks x 256 threads
  k_weights<<<512, 256, 0, stream>>>(mean, Woff, boff, wk);
  // Stage 3: tiles of 32x8 outputs
  dim3 g3(W2 / TLW, H2 / TLH, BB);            // (8, 32, 2)
  k_carafe<<<g3, 256, 0, stream>>>(x, wk, out);
}